// WaveCell_75084618268978
// MI455X (gfx1250) — compile-verified
//
#include <hip/hip_runtime.h>
#include <stdint.h>

// ---------------------------------------------------------------------------
// 2D elastic wave time step (seistorch), fused two-stage stencil.
// MI455X / gfx1250: wave32, async global->LDS b128 staging (ASYNCcnt),
// float4 (b128) global loads/stores, LDS tiles for stage-1 intermediates.
// Memory-bound: ~537 MB minimum traffic -> ~23us @ 23.3 TB/s.
// WMMA intentionally unused: 2-point f32 stencil, ~0.5 GFLOP total.
// ---------------------------------------------------------------------------

#define NZC 1024
#define NXC 2048
#define NB  4
#define TZ  32
#define TX  64
#define NTHREADS 256

#define VW 72               // staging tile width   (x: x0-4 .. x0+67), 16B aligned
#define VH 34               // staging tile height  (z: z0-1 .. z0+32)
#define SW 72               // stage-1 tile width   (x: x0-4 .. x0+67)
#define SH 33               // stage-1 tile height  (z: z0   .. z0+32)
#define NCH (VW / 4)        // 18 float4 chunks per row

#define FLD   (NZC * NXC)   // elements per field per batch
#define TOTAL (NB * FLD)    // elements per field

__device__ __forceinline__ void async_g2l_b128(const float* gptr, const float* lptr) {
    // CDNA5 async copy: 16B per lane, global -> LDS, tracked by ASYNCcnt.
    // VDST = per-lane LDS byte address (generic addr truncates to [31:0]),
    // VADDR = 64-bit global address (GV mode).
    uint32_t lds_addr = (uint32_t)(uintptr_t)lptr;
    asm volatile("global_load_async_to_lds_b128 %0, %1, off"
                 :
                 : "v"(lds_addr), "v"(gptr)
                 : "memory");
}

__device__ __forceinline__ void wait_asynccnt0() {
#if __has_builtin(__builtin_amdgcn_s_wait_asynccnt)
    __builtin_amdgcn_s_wait_asynccnt(0);
#else
    asm volatile("s_wait_asynccnt 0x0" ::: "memory");
#endif
}

__device__ __forceinline__ int clamp_off(int off) {
    off = off < 0 ? 0 : off;
    return off > (TOTAL - 4) ? (TOTAL - 4) : off;
}

__global__ __launch_bounds__(NTHREADS)
void wave_step_kernel(const float* __restrict__ vp,  const float* __restrict__ vs,
                      const float* __restrict__ rho, const float* __restrict__ p,
                      const float* __restrict__ vx,  const float* __restrict__ vz,
                      const float* __restrict__ txx, const float* __restrict__ tzz,
                      const float* __restrict__ txz, const float* __restrict__ d,
                      float* __restrict__ out)
{
    __shared__ __align__(16) float s_vx [VH * VW];
    __shared__ __align__(16) float s_vz [VH * VW];
    __shared__ __align__(16) float s_txx[SH * SW];
    __shared__ __align__(16) float s_tzz[SH * SW];
    __shared__ __align__(16) float s_txz[SH * SW];
    __shared__ __align__(16) float s_p  [SH * SW];

    const int tid = (int)threadIdx.x;
    const int x0  = (int)blockIdx.x * TX;
    const int z0  = (int)blockIdx.y * TZ;
    const int b   = (int)blockIdx.z;
    const int base = b * FLD;

    float* __restrict__ out_p   = out + 0 * (size_t)TOTAL;
    float* __restrict__ out_vx  = out + 1 * (size_t)TOTAL;
    float* __restrict__ out_vz  = out + 2 * (size_t)TOTAL;
    float* __restrict__ out_txx = out + 3 * (size_t)TOTAL;
    float* __restrict__ out_tzz = out + 4 * (size_t)TOTAL;
    float* __restrict__ out_txz = out + 5 * (size_t)TOTAL;

    const float c1 = 1.0e-4f;   // DT / H

    // ---- Stage 0: async-stage vx, vz (+halo) into LDS, b128 per lane ----
    // Chunk addresses are clamped into the buffer; any clamped chunk only
    // feeds halo cells whose diffs are analytically zeroed below.
    for (int t = tid; t < VH * NCH; t += NTHREADS) {
        int lz = t / NCH;
        int c  = t - lz * NCH;
        int gz = z0 - 1 + lz;
        gz = gz < 0 ? 0 : (gz > NZC - 1 ? NZC - 1 : gz);
        int off = clamp_off(base + gz * NXC + (x0 - 4 + 4 * c));
        int l   = lz * VW + 4 * c;
        async_g2l_b128(vx + off, &s_vx[l]);
        async_g2l_b128(vz + off, &s_vz[l]);
    }
    wait_asynccnt0();
    __syncthreads();

    // ---- Stage 1: stress/pressure update on extended region --------------
    // Rows r = 0..32 (z0..z0+32), chunks c = 0..17 (x = x0-4 .. x0+67).
    for (int t = tid; t < SH * NCH; t += NTHREADS) {
        int r  = t / NCH;
        int c  = t - r * NCH;
        int gz  = z0 + r;
        int cz  = gz > NZC - 1 ? NZC - 1 : gz;
        int gxs = x0 - 4 + 4 * c;
        int off = clamp_off(base + cz * NXC + gxs);

        float4 rho4 = *(const float4*)(rho + off);
        float4 d4   = *(const float4*)(d   + off);
        float4 vp4  = *(const float4*)(vp  + off);
        float4 vs4  = *(const float4*)(vs  + off);
        float4 p4   = *(const float4*)(p   + off);
        float4 txx4 = *(const float4*)(txx + off);
        float4 tzz4 = *(const float4*)(tzz + off);
        float4 txz4 = *(const float4*)(txz + off);

        const int lrow  = (r + 1) * VW;   // vx/vz row for gz
        const int lrowm = r * VW;         // row for gz-1
        const int lx    = 4 * c;
        float4 vxc4 = *(const float4*)&s_vx[lrow  + lx];
        float4 vzc4 = *(const float4*)&s_vz[lrow  + lx];
        float4 vxm4 = *(const float4*)&s_vx[lrowm + lx];
        float4 vzm4 = *(const float4*)&s_vz[lrowm + lx];
        int ridx = lx + 4; ridx = ridx > VW - 1 ? VW - 1 : ridx;  // c==17 pad
        int lidx = lx - 1; lidx = lidx < 0 ? 0 : lidx;            // c==0  pad
        float vxR = s_vx[lrow + ridx];
        float vzL = s_vz[lrow + lidx];

        float avx[5]  = {vxc4.x, vxc4.y, vxc4.z, vxc4.w, vxR};
        float avz[5]  = {vzL, vzc4.x, vzc4.y, vzc4.z, vzc4.w};
        float avxm[4] = {vxm4.x, vxm4.y, vxm4.z, vxm4.w};
        float avzm[4] = {vzm4.x, vzm4.y, vzm4.z, vzm4.w};
        float arho[4] = {rho4.x, rho4.y, rho4.z, rho4.w};
        float ad[4]   = {d4.x, d4.y, d4.z, d4.w};
        float avp[4]  = {vp4.x, vp4.y, vp4.z, vp4.w};
        float avs[4]  = {vs4.x, vs4.y, vs4.z, vs4.w};
        float ap[4]   = {p4.x, p4.y, p4.z, p4.w};
        float atxx[4] = {txx4.x, txx4.y, txx4.z, txx4.w};
        float atzz[4] = {tzz4.x, tzz4.y, tzz4.z, tzz4.w};
        float atxz[4] = {txz4.x, txz4.y, txz4.z, txz4.w};

        float ytxx[4], ytzz[4], ytxz[4], yp[4];
        #pragma unroll
        for (int j = 0; j < 4; ++j) {
            int gx = gxs + j;
            float cc   = 0.0005f * ad[j];        // 0.5*DT*d
            float invv = 1.0f / (1.0f + cc);
            float decv = 1.0f - cc;
            // jax roll+zero-slice semantics:
            float vx_x = (gx >= NXC - 1) ? 0.0f : avx[j + 1] - avx[j];
            float vz_z = (gz <= 0)       ? 0.0f : avz[j + 1] - avzm[j];
            float vx_z = (gz <= 0)       ? 0.0f : avx[j]     - avxm[j];
            float vz_x = (gx <= 0)       ? 0.0f : avz[j + 1] - avz[j];
            float mu  = arho[j] * avs[j] * avs[j];
            float lpm = arho[j] * (avp[j] * avp[j] - avs[j] * avs[j]); // lam+mu
            ytxx[j] = invv * ( c1 * mu  * (vx_x - vz_z) + decv * atxx[j]);
            ytzz[j] = invv * ( c1 * mu  * (vz_z - vx_x) + decv * atzz[j]);
            ytxz[j] = invv * ( c1 * mu  * (vz_x + vx_z) + decv * atxz[j]);
            yp[j]   = invv * (-c1 * lpm * (vx_x + vz_z) + decv * ap[j]);
        }

        int srow = r * SW + lx;
        *(float4*)&s_txx[srow] = make_float4(ytxx[0], ytxx[1], ytxx[2], ytxx[3]);
        *(float4*)&s_tzz[srow] = make_float4(ytzz[0], ytzz[1], ytzz[2], ytzz[3]);
        *(float4*)&s_txz[srow] = make_float4(ytxz[0], ytxz[1], ytxz[2], ytxz[3]);
        *(float4*)&s_p  [srow] = make_float4(yp[0],   yp[1],   yp[2],   yp[3]);

        // central chunks (x in [x0, x0+63], z in [z0, z0+31]) are outputs
        if (r < TZ && c >= 1 && c <= 16) {
            size_t o = (size_t)(base + gz * NXC + gxs);
            *(float4*)(out_p   + o) = make_float4(yp[0],   yp[1],   yp[2],   yp[3]);
            *(float4*)(out_txx + o) = make_float4(ytxx[0], ytxx[1], ytxx[2], ytxx[3]);
            *(float4*)(out_tzz + o) = make_float4(ytzz[0], ytzz[1], ytzz[2], ytzz[3]);
            *(float4*)(out_txz + o) = make_float4(ytxz[0], ytxz[1], ytxz[2], ytxz[3]);
        }
    }
    __syncthreads();

    // ---- Stage 2: velocity update on central tile -------------------------
    for (int t = tid; t < TZ * (TX / 4); t += NTHREADS) {   // 32*16 = 512
        int zr = t >> 4;
        int cq = t & 15;
        int gz  = z0 + zr;
        int gxs = x0 + 4 * cq;
        int off = base + gz * NXC + gxs;

        float4 rho4 = *(const float4*)(rho + off);
        float4 d4   = *(const float4*)(d   + off);

        const int sc  = 4 * cq + 4;           // stage-1 col of gxs
        const int sr0 = zr * SW + sc;
        const int sr1 = (zr + 1) * SW + sc;
        float4 txxc = *(const float4*)&s_txx[sr0];
        float4 tzzc = *(const float4*)&s_tzz[sr0];
        float4 txzc = *(const float4*)&s_txz[sr0];
        float4 pc4  = *(const float4*)&s_p  [sr0];
        float4 tzz1 = *(const float4*)&s_tzz[sr1];
        float4 txz1 = *(const float4*)&s_txz[sr1];
        float4 p1   = *(const float4*)&s_p  [sr1];
        float txxL = s_txx[sr0 - 1];
        float pL   = s_p  [sr0 - 1];
        float txzL = s_txz[sr0 - 1];
        float txzR = s_txz[sr0 + 4];

        const int vrow = (zr + 1) * VW + (4 * cq + 4);
        float4 vx4 = *(const float4*)&s_vx[vrow];
        float4 vz4 = *(const float4*)&s_vz[vrow];

        float atxx[5] = {txxL, txxc.x, txxc.y, txxc.z, txxc.w};
        float ap[5]   = {pL,   pc4.x,  pc4.y,  pc4.z,  pc4.w};
        float atxz[6] = {txzL, txzc.x, txzc.y, txzc.z, txzc.w, txzR};
        float atzzc[4]= {tzzc.x, tzzc.y, tzzc.z, tzzc.w};
        float atzz1[4]= {tzz1.x, tzz1.y, tzz1.z, tzz1.w};
        float atxz1[4]= {txz1.x, txz1.y, txz1.z, txz1.w};
        float ap1[4]  = {p1.x,  p1.y,  p1.z,  p1.w};
        float arho[4] = {rho4.x, rho4.y, rho4.z, rho4.w};
        float ad[4]   = {d4.x, d4.y, d4.z, d4.w};
        float avx[4]  = {vx4.x, vx4.y, vx4.z, vx4.w};
        float avz[4]  = {vz4.x, vz4.y, vz4.z, vz4.w};

        const bool zlast = (gz == NZC - 1);
        float yvx[4], yvz[4];
        #pragma unroll
        for (int j = 0; j < 4; ++j) {
            int gx = gxs + j;
            float cc   = 0.0005f * ad[j];
            float invv = 1.0f / (1.0f + cc);
            float decv = 1.0f - cc;
            float txx_x = (gx == 0)       ? 0.0f : atxx[j + 1] - atxx[j];
            float p_x   = (gx == 0)       ? 0.0f : ap[j + 1]   - ap[j];
            float txz_x = (gx == NXC - 1) ? 0.0f : atxz[j + 2] - atxz[j + 1];
            float txz_z = zlast ? 0.0f : atxz1[j] - atxz[j + 1];
            float tzz_z = zlast ? 0.0f : atzz1[j] - atzzc[j];
            float p_z   = zlast ? 0.0f : ap1[j]   - ap[j + 1];
            float k = c1 / arho[j];              // DT/(rho*H)
            yvx[j] = invv * (k * (txx_x + txz_z - p_x) + decv * avx[j]);
            yvz[j] = invv * (k * (txz_x + tzz_z - p_z) + decv * avz[j]);
        }

        *(float4*)(out_vx + (size_t)off) = make_float4(yvx[0], yvx[1], yvx[2], yvx[3]);
        *(float4*)(out_vz + (size_t)off) = make_float4(yvz[0], yvz[1], yvz[2], yvz[3]);
    }
}

extern "C" void kernel_launch(void* const* d_in, const int* in_sizes, int n_in,
                              void* d_out, int out_size, void* d_ws, size_t ws_size,
                              hipStream_t stream) {
    (void)in_sizes; (void)n_in; (void)out_size; (void)d_ws; (void)ws_size;
    const float* vp  = (const float*)d_in[0];
    const float* vs  = (const float*)d_in[1];
    const float* rho = (const float*)d_in[2];
    const float* p   = (const float*)d_in[3];
    const float* vx  = (const float*)d_in[4];
    const float* vz  = (const float*)d_in[5];
    const float* txx = (const float*)d_in[6];
    const float* tzz = (const float*)d_in[7];
    const float* txz = (const float*)d_in[8];
    const float* d   = (const float*)d_in[9];
    float* out = (float*)d_out;

    dim3 grid(NXC / TX, NZC / TZ, NB);   // 32 x 32 x 4 = 4096 workgroups
    dim3 block(NTHREADS);                // 8 wave32 per WG
    wave_step_kernel<<<grid, block, 0, stream>>>(vp, vs, rho, p, vx, vz,
                                                 txx, tzz, txz, d, out);
}